// GraphConv_65300682768935
// MI455X (gfx1250) — compile-verified
//
#include <hip/hip_runtime.h>
#include <hip/hip_bf16.h>

typedef __attribute__((ext_vector_type(2))) float v2f;
typedef __attribute__((ext_vector_type(8))) float v8f;

#define FEATS 128

// ---------------------------------------------------------------- zero scratch
__global__ __launch_bounds__(256) void zero_kernel(float* p, int n) {
    int i = blockIdx.x * 256 + threadIdx.x;
    if (i < n) p[i] = 0.0f;
}

// ---------------------------------------------------------------- degree count
__global__ __launch_bounds__(256) void degree_kernel(const int* __restrict__ src,
                                                     const int* __restrict__ dst,
                                                     float* __restrict__ deg_out,
                                                     float* __restrict__ deg_in,
                                                     int n_edges) {
    int e = blockIdx.x * 256 + threadIdx.x;
    if (e < n_edges) {
        atomicAdd(&deg_out[src[e]], 1.0f);
        atomicAdd(&deg_in[dst[e]], 1.0f);
    }
}

// ---------------------------------------------------------------- norm factors
__global__ __launch_bounds__(256) void norm_kernel(const float* __restrict__ deg_out,
                                                   const float* __restrict__ deg_in,
                                                   float* __restrict__ norm,
                                                   int n_nodes) {
    int i = blockIdx.x * 256 + threadIdx.x;
    if (i < n_nodes) {
        float di = fmaxf(deg_in[i], 1.0f);
        float dz = fmaxf(deg_out[i], 1.0f);
        // deg_in^-0.5 * deg_out^-0.5
        norm[i] = rsqrtf(di) * rsqrtf(dz);
    }
}

// ---------------------------------------------------------------- edge SpMM
// one wave32 per edge; lane l handles features [4l, 4l+3] (float4 = 512B/wave)
__global__ __launch_bounds__(256) void aggregate_kernel(const float* __restrict__ feat,
                                                        const int* __restrict__ src,
                                                        const int* __restrict__ dst,
                                                        const float* __restrict__ norm,
                                                        float* __restrict__ agg,
                                                        int n_edges) {
    int wid  = threadIdx.x >> 5;
    int lane = threadIdx.x & 31;
    int e = blockIdx.x * 8 + wid;
    if (e >= n_edges) return;

    int s = src[e];
    int d = dst[e];
    float nrm = norm[s];

    const float4* fp = (const float4*)(feat + (size_t)s * FEATS);
    float4 v = fp[lane];

    float* ap = agg + (size_t)d * FEATS + lane * 4;
    atomicAdd(ap + 0, v.x * nrm);
    atomicAdd(ap + 1, v.y * nrm);
    atomicAdd(ap + 2, v.z * nrm);
    atomicAdd(ap + 3, v.w * nrm);
}

// ---------------------------------------------------------------- WMMA GEMM
// out[50000x128] = relu(agg[50000x128] @ W[128x128] + bias)
// one wave per 16x16 tile; block = 8 waves = the 8 N-tiles of one M-tile.
// fp32 WMMA 16x16x4, chained 32x along K=128.
__global__ __launch_bounds__(256) void gemm_bias_relu_kernel(const float* __restrict__ agg,
                                                             const float* __restrict__ W,
                                                             const float* __restrict__ bias,
                                                             float* __restrict__ out) {
    const int lane = threadIdx.x & 31;
    const int wave = threadIdx.x >> 5;
    const int m0 = blockIdx.x * 16;     // 3125 blocks * 16 = 50000 exactly
    const int n0 = wave * 16;           // 8 waves * 16 = 128 cols

    const int rc = lane & 15;           // A row / B col index for this lane
    const int kh = (lane >> 4) * 2;     // K sub-offset: 0 for lanes 0-15, 2 for 16-31

    // A: row (m0+rc), elements k+kh, k+kh+1 are contiguous (row-major)
    const float* Arow = agg + (size_t)(m0 + rc) * FEATS + kh;
    // B: col (n0+rc), rows k+kh, k+kh+1 (row-major, stride FEATS)
    const float* Bcol = W + (size_t)kh * FEATS + n0 + rc;

    v8f c = {};
    #pragma unroll
    for (int k = 0; k < FEATS; k += 4) {
        v2f a;
        a.x = Arow[k];
        a.y = Arow[k + 1];
        v2f b;
        b.x = Bcol[(size_t)k * FEATS];
        b.y = Bcol[(size_t)(k + 1) * FEATS];
        // D = A(16x4) * B(4x16) + C  -- full fp32 matrix pipe
        c = __builtin_amdgcn_wmma_f32_16x16x4_f32(false, a, false, b,
                                                  (short)0, c, false, false);
    }

    // C/D layout: VGPR i -> M = i (lanes 0-15) or 8+i (lanes 16-31); N = rc
    const int n = n0 + rc;
    const float bv = bias[n];
    const int mbase = m0 + ((lane >> 4) << 3);
    #pragma unroll
    for (int i = 0; i < 8; ++i) {
        float v = c[i] + bv;
        out[(size_t)(mbase + i) * FEATS + n] = fmaxf(v, 0.0f);
    }
}

// ---------------------------------------------------------------- launch
extern "C" void kernel_launch(void* const* d_in, const int* in_sizes, int n_in,
                              void* d_out, int out_size, void* d_ws, size_t ws_size,
                              hipStream_t stream) {
    const float* feat   = (const float*)d_in[0];
    const float* weight = (const float*)d_in[1];
    const float* bias   = (const float*)d_in[2];
    const int*   src    = (const int*)d_in[3];
    const int*   dst    = (const int*)d_in[4];
    float*       out    = (float*)d_out;

    const int n_nodes = in_sizes[0] / FEATS;   // 50000
    const int n_edges = in_sizes[3];           // 800000

    // workspace layout (floats)
    float* deg_out = (float*)d_ws;
    float* deg_in  = deg_out + n_nodes;
    float* norm    = deg_in + n_nodes;
    float* agg     = norm + n_nodes;           // offset 150000 floats (16B aligned)

    const int zero_count = 3 * n_nodes + n_nodes * FEATS;  // degs + norm + agg
    zero_kernel<<<(zero_count + 255) / 256, 256, 0, stream>>>((float*)d_ws, zero_count);

    degree_kernel<<<(n_edges + 255) / 256, 256, 0, stream>>>(src, dst, deg_out, deg_in, n_edges);

    norm_kernel<<<(n_nodes + 255) / 256, 256, 0, stream>>>(deg_out, deg_in, norm, n_nodes);

    // 8 edges (waves) per 256-thread block
    aggregate_kernel<<<(n_edges + 7) / 8, 256, 0, stream>>>(feat, src, dst, norm, agg, n_edges);

    // 16-row M tiles; n_nodes = 50000 is an exact multiple of 16
    gemm_bias_relu_kernel<<<n_nodes / 16, 256, 0, stream>>>(agg, weight, bias, out);
}